// _fasterRCNN_73435350827079
// MI455X (gfx1250) — compile-verified
//
#include <hip/hip_runtime.h>
#include <math.h>

typedef __attribute__((ext_vector_type(2))) float v2f;
typedef __attribute__((ext_vector_type(8))) float v8f;

#define NTGT 2000
#define NSRC 2000
#define NCLS 101
#define KBOX 64
#define HF 75
#define WF 125
#define NPIX (HF * WF)

// workspace layout (units of 4 bytes)
#define DM_OFF   0      // 101 floats: delta_margin accumulators
#define NPC_OFF  101    // 101 floats: num_per_class accumulators
#define LOSS_OFF 202    // 4 floats: loss sums
#define NW_OFF   208    // 101 floats: norm_w
#define GP_OFF   320    // 64 ints: packed gt boxes
#define GV_OFF   384    // 64 floats: gt box values
#define PL_OFF   512    // 2000 ints: pl (argmax class per tgt row)
#define TP_OFF   2560   // 2000 ints: packed tgt boxes (reversed order)
#define TV_OFF   4560   // 2000 floats: tgt box values (reversed order)

// output layout
#define O_LOSS 0
#define O_TMF  4
#define O_CSW  105
#define O_WAD  2105
#define O_TWAD 11480

__device__ __forceinline__ int clamp127(float v) {
    int x = (int)floorf(v);
    return x < 0 ? 0 : (x > 127 ? 127 : x);
}

// ---------------- init: zero accumulators ----------------
__global__ void k_init(float* wsf) {
    int t = threadIdx.x;
    if (t < 206) wsf[t] = 0.0f;
}

// ---------------- kernel 1: softmax stats, one wave per 16 rows ----------------
// A-matrix 16x4 f32 layout: lane L holds row L%16; VGPR0=K(0|2), VGPR1=K(1|3).
// B = ones(4x16)  =>  D[m][n] = chunk row-sum(m); accumulate over 26 chunks via C.
__global__ void k_row(const float* __restrict__ score, float* __restrict__ wsf,
                      int* __restrict__ wsi) {
    int gtid = blockIdx.x * blockDim.x + threadIdx.x;
    int wave = gtid >> 5;
    int lane = threadIdx.x & 31;
    int r0 = wave * 16;
    if (r0 >= NTGT) return;              // uniform per wave (2000 % 16 == 0)
    int half = lane >> 4;
    int m = lane & 15;
    const float* rowp = score + (size_t)(r0 + m) * NCLS;

    // pass 1: row max + first-occurrence argmax
    float bestv = -1e30f; int besti = 0;
    for (int c = 0; c < 26; ++c) {
        int k0 = c * 4 + half * 2;
        float v0 = (k0 < NCLS) ? rowp[k0] : -1e30f;
        float v1 = (k0 + 1 < NCLS) ? rowp[k0 + 1] : -1e30f;
        if (v0 > bestv) { bestv = v0; besti = k0; }
        if (v1 > bestv) { bestv = v1; besti = k0 + 1; }
    }
    float ov = __shfl_xor(bestv, 16, 32);
    int   oi = __shfl_xor(besti, 16, 32);
    if (ov > bestv || (ov == bestv && oi < besti)) { bestv = ov; besti = oi; }

    // pass 2: rowsum of exp(s - max) via matrix pipe
    float rowsum;
#if defined(__gfx1250__) && __has_builtin(__builtin_amdgcn_wmma_f32_16x16x4_f32)
    v2f b; b.x = 1.0f; b.y = 1.0f;
    v8f acc = {};
    #pragma unroll
    for (int c = 0; c < 26; ++c) {
        int k0 = c * 4 + half * 2;
        float e0 = (k0 < NCLS) ? expf(rowp[k0] - bestv) : 0.0f;
        float e1 = (k0 + 1 < NCLS) ? expf(rowp[k0 + 1] - bestv) : 0.0f;
        v2f a; a.x = e0; a.y = e1;
        acc = __builtin_amdgcn_wmma_f32_16x16x4_f32(false, a, false, b,
                                                    (short)0, acc, false, false);
    }
    // D layout: lanes 0-15 hold rows 0-7 in acc[0..7]; lanes 16-31 hold rows 8-15.
    int idx = lane & 7;                  // = (L%16) & 7
    float tA = acc[0];
    tA = (idx == 1) ? acc[1] : tA;
    tA = (idx == 2) ? acc[2] : tA;
    tA = (idx == 3) ? acc[3] : tA;
    tA = (idx == 4) ? acc[4] : tA;
    tA = (idx == 5) ? acc[5] : tA;
    tA = (idx == 6) ? acc[6] : tA;
    tA = (idx == 7) ? acc[7] : tA;
    float oth = __shfl_xor(tA, 16, 32);
    bool hasLocal = ((lane >= 16) == ((lane & 8) != 0));
    rowsum = hasLocal ? tA : oth;
#else
    float local = 0.0f;
    for (int c = 0; c < 26; ++c) {
        int k0 = c * 4 + half * 2;
        if (k0 < NCLS)     local += expf(rowp[k0] - bestv);
        if (k0 + 1 < NCLS) local += expf(rowp[k0 + 1] - bestv);
    }
    rowsum = local + __shfl_xor(local, 16, 32);
#endif

    float pmax = 1.0f / rowsum;   // exp(max - max)/sum
    float margin = pmax - (1.0f - pmax) * (1.0f / (float)(NCLS - 1));
    if (lane < 16) {
        wsi[PL_OFF + r0 + m] = besti;
        if (besti != 0) {
            atomicAdd(&wsf[DM_OFF + besti], margin);
            atomicAdd(&wsf[NPC_OFF + besti], 1.0f);
        }
    }
}

// ---------------- kernel 2: weight vector, normalization, gt-box prep ----------------
__global__ void k_weights(const float* __restrict__ tmf, const float* __restrict__ records,
                          const float* __restrict__ gt_boxes, const int* __restrict__ h1,
                          const int* __restrict__ w1, float* __restrict__ wsf,
                          int* __restrict__ wsi, float* __restrict__ out) {
    __shared__ float sred[128];
    __shared__ float smin[128];
    __shared__ float smax[128];
    __shared__ float snorm[NCLS];
    int t = threadIdx.x;
    float rec = records[0];
    float np = (t < NCLS) ? wsf[NPC_OFF + t] : 0.0f;
    sred[t] = np;
    __syncthreads();
    for (int s = 64; s > 0; s >>= 1) { if (t < s) sred[t] += sred[t + s]; __syncthreads(); }
    float numsum = sred[0];

    float dm = (t < NCLS) ? wsf[DM_OFF + t] : 0.0f;
    float tm = (t < NCLS) ? tmf[t] : 0.0f;
    float wv = (numsum > 0.0f) ? (tm * rec + dm / (np + 1e-6f)) / (rec + 1.0f) : tm;
    if (t < NCLS) out[O_TMF + t] = wv;        // new_tmf output

    float wc = fmaxf(wv, 0.0f);
    smin[t] = (t < NCLS) ? wc : 1e30f;
    smax[t] = (t < NCLS) ? wc : -1e30f;
    __syncthreads();
    for (int s = 64; s > 0; s >>= 1) {
        if (t < s) {
            smin[t] = fminf(smin[t], smin[t + s]);
            smax[t] = fmaxf(smax[t], smax[t + s]);
        }
        __syncthreads();
    }
    float lo = smin[0], hi = smax[0];
    float nw = (wc - lo) / (hi - lo + 1e-6f);
    if (t < NCLS) { wsf[NW_OFF + t] = nw; snorm[t] = nw; }
    __syncthreads();

    if (t < KBOX) {
        float sw = (float)WF / (float)w1[0];
        float sh = (float)HF / (float)h1[0];
        const float* g = gt_boxes + t * 5;
        int x1 = clamp127(g[0] * sw);
        int y1 = clamp127(g[1] * sh);
        int x2 = clamp127(g[2] * sw);
        int y2 = clamp127(g[3] * sh);
        int cls = (int)g[4];
        int valid = (cls != 0) ? 1 : 0;
        wsi[GP_OFF + t] = x1 | (y1 << 7) | (x2 << 14) | (y2 << 21) | (valid << 28);
        wsf[GV_OFF + t] = snorm[cls];
    }
}

// ---------------- kernel 3: prep reversed tgt boxes (reference valid-mask quirk kept) ----------------
__global__ void k_tgtprep(const float* __restrict__ tgt_rois, const int* __restrict__ h1,
                          const int* __restrict__ w1, float* __restrict__ wsf,
                          int* __restrict__ wsi) {
    int i = blockIdx.x * blockDim.x + threadIdx.x;
    if (i >= NTGT) return;
    int j = NTGT - 1 - i;                         // reversed index
    float sw = (float)WF / (float)w1[0];
    float sh = (float)HF / (float)h1[0];
    const float* r = tgt_rois + (size_t)j * 5;    // [batch_idx, x1, y1, x2, y2]
    int x1 = clamp127(r[1] * sw);
    int y1 = clamp127(r[2] * sh);
    int x2 = clamp127(r[3] * sw);
    int y2 = clamp127(r[4] * sh);
    int valid = (wsi[PL_OFF + i] != 0) ? 1 : 0;   // NOTE: un-reversed pl, per reference
    wsi[TP_OFF + i] = x1 | (y1 << 7) | (x2 << 14) | (y2 << 21) | (valid << 28);
    wsf[TV_OFF + i] = wsf[NW_OFF + wsi[PL_OFF + j]];   // reversed pl for value
}

// ---------------- kernel 4: paint gt boxes + da_img partial sum ----------------
__global__ void k_paint_gt(const float* __restrict__ base_score, float* __restrict__ wsf,
                           const int* __restrict__ wsi, float* __restrict__ out) {
    int p = blockIdx.x * blockDim.x + threadIdx.x;
    float term = 0.0f;
    if (p < NPIX) {
        int y = p / WF, x = p - y * WF;
        float wad = 0.0f;
        for (int i = KBOX - 1; i >= 0; --i) {     // last-write-wins => highest index first
            int pk = wsi[GP_OFF + i];
            if (pk >> 28) {
                int x1 = pk & 127, y1 = (pk >> 7) & 127;
                int x2 = (pk >> 14) & 127, y2 = (pk >> 21) & 127;
                if (x >= x1 && x <= x2 && y >= y1 && y <= y2) {
                    wad = wsf[GV_OFF + i];
                    break;
                }
            }
        }
        out[O_WAD + p] = wad;
        term = (1.0f + wad) * (-logf(base_score[p]));   // t=1 BCE term
    }
    for (int o = 16; o >= 1; o >>= 1) term += __shfl_xor(term, o, 32);
    if ((threadIdx.x & 31) == 0) atomicAdd(&wsf[LOSS_OFF + 0], term);
}

// ---------------- kernel 5: paint tgt boxes + tgt_da_img partial sum ----------------
__global__ void k_paint_tgt(const float* __restrict__ tgt_base_score, float* __restrict__ wsf,
                            const int* __restrict__ wsi, float* __restrict__ out) {
    int p = blockIdx.x * blockDim.x + threadIdx.x;
    float term = 0.0f;
    if (p < NPIX) {
        int y = p / WF, x = p - y * WF;
        float twad = 0.0f;
        for (int i = NTGT - 1; i >= 0; --i) {
            int pk = wsi[TP_OFF + i];
            if (pk >> 28) {
                int x1 = pk & 127, y1 = (pk >> 7) & 127;
                int x2 = (pk >> 14) & 127, y2 = (pk >> 21) & 127;
                if (x >= x1 && x <= x2 && y >= y1 && y <= y2) {
                    twad = wsf[TV_OFF + i];
                    break;
                }
            }
        }
        out[O_TWAD + p] = twad;
        term = (1.0f + twad) * (-logf(1.0f - tgt_base_score[p]));   // t=0 BCE term
    }
    for (int o = 16; o >= 1; o >>= 1) term += __shfl_xor(term, o, 32);
    if ((threadIdx.x & 31) == 0) atomicAdd(&wsf[LOSS_OFF + 2], term);
}

// ---------------- kernel 6: instance losses + common_source_weight ----------------
__global__ void k_inst(const float* __restrict__ inst, const float* __restrict__ tinst,
                       const int* __restrict__ rlab, float* __restrict__ wsf,
                       const int* __restrict__ wsi, float* __restrict__ out) {
    int i = blockIdx.x * blockDim.x + threadIdx.x;
    float t1 = 0.0f, t2 = 0.0f;
    if (i < NSRC) {
        float csw = wsf[NW_OFF + rlab[i]];
        out[O_CSW + i] = csw;
        t1 = (1.0f + csw) * (-logf(inst[i]));
    }
    if (i < NTGT) {
        float nwp = wsf[NW_OFF + wsi[PL_OFF + i]];
        t2 = (1.0f + nwp) * (-logf(1.0f - tinst[i]));
    }
    for (int o = 16; o >= 1; o >>= 1) {
        t1 += __shfl_xor(t1, o, 32);
        t2 += __shfl_xor(t2, o, 32);
    }
    if ((threadIdx.x & 31) == 0) {
        atomicAdd(&wsf[LOSS_OFF + 1], t1);
        atomicAdd(&wsf[LOSS_OFF + 3], t2);
    }
}

// ---------------- kernel 7: finalize losses ----------------
__global__ void k_fin(const float* __restrict__ wsf, float* __restrict__ out) {
    int t = threadIdx.x;
    if (t == 0) out[0] = wsf[LOSS_OFF + 0] / (float)NPIX;
    if (t == 1) out[1] = wsf[LOSS_OFF + 1] / (float)NSRC;
    if (t == 2) out[2] = wsf[LOSS_OFF + 2] / (float)NPIX;
    if (t == 3) out[3] = wsf[LOSS_OFF + 3] / (float)NTGT;
}

extern "C" void kernel_launch(void* const* d_in, const int* in_sizes, int n_in,
                              void* d_out, int out_size, void* d_ws, size_t ws_size,
                              hipStream_t stream) {
    (void)in_sizes; (void)n_in; (void)out_size; (void)ws_size;
    const float* tgt_cls_score = (const float*)d_in[0];
    const float* base_score    = (const float*)d_in[1];
    const float* tgt_base      = (const float*)d_in[2];
    const float* inst_sig      = (const float*)d_in[3];
    const float* tgt_inst_sig  = (const float*)d_in[4];
    const float* tmf           = (const float*)d_in[5];
    const float* records       = (const float*)d_in[6];
    const float* gt_boxes      = (const float*)d_in[7];
    const float* tgt_rois      = (const float*)d_in[8];
    const int*   rois_label    = (const int*)d_in[9];
    const int*   h1            = (const int*)d_in[10];
    const int*   w1            = (const int*)d_in[11];
    float* out = (float*)d_out;
    float* wsf = (float*)d_ws;
    int*   wsi = (int*)d_ws;

    k_init<<<1, 256, 0, stream>>>(wsf);
    // 2000 rows / 16 per wave = 125 waves -> 16 blocks x 8 waves
    k_row<<<16, 256, 0, stream>>>(tgt_cls_score, wsf, wsi);
    k_weights<<<1, 128, 0, stream>>>(tmf, records, gt_boxes, h1, w1, wsf, wsi, out);
    k_tgtprep<<<(NTGT + 255) / 256, 256, 0, stream>>>(tgt_rois, h1, w1, wsf, wsi);
    k_paint_gt<<<(NPIX + 255) / 256, 256, 0, stream>>>(base_score, wsf, wsi, out);
    k_paint_tgt<<<(NPIX + 255) / 256, 256, 0, stream>>>(tgt_base, wsf, wsi, out);
    k_inst<<<(NTGT + 255) / 256, 256, 0, stream>>>(inst_sig, tgt_inst_sig, rois_label, wsf, wsi, out);
    k_fin<<<1, 32, 0, stream>>>(wsf, out);
}